// Same5_58875411693684
// MI455X (gfx1250) — compile-verified
//
#include <hip/hip_runtime.h>
#include <math.h>

typedef __attribute__((ext_vector_type(2))) float v2f;
typedef __attribute__((ext_vector_type(8))) float v8f;

#define KDIM 256
#define DDIM 256
#define PDIM 40
#define CDIM 32
#define EPS_NORM 1e-12f

#define BM 64
#define BN 64
#define KC 32
#define LDT 33   // padded LDS row stride (floats) to dodge bank conflicts

// Fast tanh on the TRANS units: v_exp_f32 + v_rcp_f32 instead of ocml's
// divide-based tanhf.  tanh(x) = sign(x) * (1 - 2/(exp(2|x|)+1)).
// 2*log2(e) = 2.8853900817779268
__device__ __forceinline__ float fast_tanh(float x) {
    float ax = fabsf(x);
    float e  = __builtin_amdgcn_exp2f(ax * 2.88539008f);   // exp(2|x|)
    float r  = 1.0f - 2.0f * __builtin_amdgcn_rcpf(e + 1.0f);
    return copysignf(r, x);
}

// ---------------- zero accumulators ----------------
__global__ void zero_kernel(float* __restrict__ buf, int n) {
    int i = blockIdx.x * blockDim.x + threadIdx.x;
    if (i < n) buf[i] = 0.0f;
}

// ---------------- normalize + transpose ----------------
// in: [K, D, P]  ->  out: [P, K, D], out[p][k][d] = in[k][d][p] / max(||in[k,:,p]||2, eps)
__global__ void __launch_bounds__(256)
normalize_kernel(const float* __restrict__ enroll,
                 const float* __restrict__ testm,
                 float* __restrict__ wsA,
                 float* __restrict__ wsB) {
    __shared__ float tile[DDIM * PDIM];   // 40 KB (of 320 KB/WGP)
    __shared__ float rnorm[PDIM];

    const int bx = blockIdx.x;
    const float* in  = (bx < KDIM) ? enroll : testm;
    float*       out = (bx < KDIM) ? wsA    : wsB;
    const int k = (bx < KDIM) ? bx : bx - KDIM;

    const float* src = in + (size_t)k * DDIM * PDIM;
    for (int i = threadIdx.x; i < DDIM * PDIM; i += 256)
        tile[i] = src[i];                       // coalesced, layout [d][p]
    __syncthreads();

    if (threadIdx.x < PDIM) {
        const int p = threadIdx.x;
        float s = 0.0f;
        for (int d = 0; d < DDIM; ++d) {
            float v = tile[d * PDIM + p];
            s = fmaf(v, v, s);
        }
        rnorm[p] = 1.0f / fmaxf(sqrtf(s), EPS_NORM);
    }
    __syncthreads();

    for (int i = threadIdx.x; i < DDIM * PDIM; i += 256) {
        const int p = i >> 8;     // / 256
        const int d = i & 255;    // % 256
        out[((size_t)p * KDIM + k) * DDIM + d] = tile[d * PDIM + p] * rnorm[p];
    }
}

// ---------------- fused per-phoneme GEMM + tanh-MLP epilogue ----------------
// wsA/wsB: [P][K][D] normalized.  For p = blockIdx.z compute a 64x64 tile of
// S_p = A_p * B_p^T (contraction over D) entirely in WMMA accumulators, then
// apply  sw = sum_c fc2[c]*tanh(s*fc1w[c]+fc1b[c]);  swp = tanh(sw*s)
// and atomically accumulate num += swp*w_p,  den += (s!=0)*w_p  over p.
__global__ void __launch_bounds__(256)
gemm_epilogue_kernel(const float* __restrict__ wsA,
                     const float* __restrict__ wsB,
                     const float* __restrict__ pho_weight,
                     const float* __restrict__ fc1_w,
                     const float* __restrict__ fc1_b,
                     const float* __restrict__ fc2_w,
                     float* __restrict__ num,
                     float* __restrict__ den) {
    __shared__ float As[BM * LDT];
    __shared__ float Bs[BN * LDT];
    __shared__ float sW1[CDIM], sB1[CDIM], sW2[CDIM];

    const int tid  = threadIdx.x;
    const int lane = tid & 31;
    const int wave = tid >> 5;
    const int wm   = wave >> 1;   // 0..3 : 16-row strip within 64
    const int wn   = wave & 1;    // 0..1 : 32-col strip within 64
    const int ta = blockIdx.x, tb = blockIdx.y, p = blockIdx.z;

    if (tid < CDIM) { sW1[tid] = fc1_w[tid]; sB1[tid] = fc1_b[tid]; sW2[tid] = fc2_w[tid]; }

    // positive_weight[p] = (w[p]-min)/(1e-6+max-min), ALPHA=1 (tiny, L2-resident)
    float wmn = pho_weight[0], wmx = pho_weight[0];
    for (int i = 1; i < PDIM; ++i) {
        float v = pho_weight[i];
        wmn = fminf(wmn, v); wmx = fmaxf(wmx, v);
    }
    const float w_p = (pho_weight[p] - wmn) / (1e-6f + wmx - wmn);

    const float* Ap = wsA + ((size_t)p * KDIM + ta * BM) * DDIM;
    const float* Bp = wsB + ((size_t)p * KDIM + tb * BN) * DDIM;

    v8f acc0 = {};
    v8f acc1 = {};

    // V_WMMA_F32_16X16X4_F32 fragment addressing (ISA 7.12.2):
    //   A 16x4: lanes 0-15 -> {A[M=l][K=k],A[M=l][K=k+1]}, lanes 16-31 -> K=k+2,k+3
    //   B 4x16: lanes 0-15 -> {B[K=k][N=l],B[K=k+1][N=l]}, lanes 16-31 -> K=k+2,k+3
    // With Bs stored as [col][kk] (= B^T) both fragments are contiguous float2s.
    const int hi    = (lane >> 4) & 1;
    const int l15   = lane & 15;
    const int arow  = wm * 16 + l15;
    const int bcol0 = wn * 32 + l15;
    const int bcol1 = bcol0 + 16;

    for (int k0 = 0; k0 < DDIM; k0 += KC) {
        __syncthreads();
        for (int i = tid; i < BM * KC; i += 256) {
            int r = i >> 5, c = i & 31;                       // 32 consecutive cols/row -> coalesced
            As[r * LDT + c] = Ap[(size_t)r * DDIM + k0 + c];
        }
        for (int i = tid; i < BN * KC; i += 256) {
            int r = i >> 5, c = i & 31;
            Bs[r * LDT + c] = Bp[(size_t)r * DDIM + k0 + c];
        }
        __syncthreads();

        #pragma unroll
        for (int kk = 0; kk < KC; kk += 4) {
            const int kf = kk + 2 * hi;
            v2f af; af.x = As[arow * LDT + kf];  af.y = As[arow * LDT + kf + 1];
            v2f b0; b0.x = Bs[bcol0 * LDT + kf]; b0.y = Bs[bcol0 * LDT + kf + 1];
            v2f b1; b1.x = Bs[bcol1 * LDT + kf]; b1.y = Bs[bcol1 * LDT + kf + 1];
            acc0 = __builtin_amdgcn_wmma_f32_16x16x4_f32(false, af, false, b0,
                                                         (short)0, acc0, false, false);
            acc1 = __builtin_amdgcn_wmma_f32_16x16x4_f32(false, af, false, b1,
                                                         (short)0, acc1, false, false);
        }
    }
    __syncthreads();

    // C/D layout: VGPR j -> M = j (lanes 0-15) / j+8 (lanes 16-31), N = lane&15.
    float ac[2][8];
    #pragma unroll
    for (int j = 0; j < 8; ++j) { ac[0][j] = acc0[j]; ac[1][j] = acc1[j]; }

    #pragma unroll
    for (int j = 0; j < 8; ++j) {
        const int a = ta * BM + wm * 16 + j + 8 * hi;
        #pragma unroll
        for (int n = 0; n < 2; ++n) {
            const float s = ac[n][j];
            const int b = tb * BN + wn * 32 + n * 16 + l15;
            float sw = 0.0f;
            #pragma unroll
            for (int c = 0; c < CDIM; ++c)
                sw = fmaf(sW2[c], fast_tanh(fmaf(s, sW1[c], sB1[c])), sw);
            const float swp = fast_tanh(sw * s);
            atomicAdd(&num[a * KDIM + b], swp * w_p);
            atomicAdd(&den[a * KDIM + b], (s != 0.0f) ? w_p : 0.0f);
        }
    }
}

// ---------------- finalize: score = num / (den + 1e-6) ----------------
__global__ void finalize_kernel(const float* __restrict__ num,
                                const float* __restrict__ den,
                                float* __restrict__ out) {
    int i = blockIdx.x * blockDim.x + threadIdx.x;
    if (i < KDIM * KDIM) out[i] = num[i] / (den[i] + 1e-6f);
}

extern "C" void kernel_launch(void* const* d_in, const int* in_sizes, int n_in,
                              void* d_out, int out_size, void* d_ws, size_t ws_size,
                              hipStream_t stream) {
    const float* enroll = (const float*)d_in[0];
    const float* testm  = (const float*)d_in[1];
    const float* pho_w  = (const float*)d_in[2];
    const float* fc1_w  = (const float*)d_in[3];
    const float* fc1_b  = (const float*)d_in[4];
    const float* fc2_w  = (const float*)d_in[5];

    const size_t mat = (size_t)PDIM * KDIM * DDIM;   // 2,621,440 floats
    float* wsA = (float*)d_ws;
    float* wsB = wsA + mat;
    float* num = wsB + mat;
    float* den = num + (size_t)KDIM * KDIM;

    zero_kernel<<<(2 * KDIM * KDIM + 255) / 256, 256, 0, stream>>>(num, 2 * KDIM * KDIM);
    normalize_kernel<<<2 * KDIM, 256, 0, stream>>>(enroll, testm, wsA, wsB);

    dim3 grid(KDIM / BM, KDIM / BN, PDIM);           // (4, 4, 40)
    gemm_epilogue_kernel<<<grid, 256, 0, stream>>>(wsA, wsB, pho_w, fc1_w, fc1_b,
                                                   fc2_w, num, den);

    finalize_kernel<<<(KDIM * KDIM + 255) / 256, 256, 0, stream>>>(num, den, (float*)d_out);
}